// STGCN_71494025609303
// MI455X (gfx1250) — compile-verified
//
#include <hip/hip_runtime.h>
#include <cstddef>

// ST-GCN for MI455X (gfx1250, wave32).
// - All dense matmuls on matrix cores via V_WMMA_F32_16X16X4_F32 (fp32 WMMA:
//   GEMM FLOPs ~10G are trivial vs the L2-resident edge scatter, so no reason
//   to drop precision).
// - All weight (B) matrices are pre-packed into K-pair layout
//   BP[(kg/2)*128 + co*2 + (kg&1)] so each lane's B fragment is one aligned
//   8B load and a 4-tile group reads a contiguous 1KB segment.
// - Graph aggregation: 1M edges x 64ch f32 atomics (h = 32MB, L2-resident).
// - BN stats reduced in LDS (ds_add) then merged with global atomics.
//
// Workspace: 3*N*64 + N + 256 + (96+32+32)*128 floats (~97 MB).

typedef float v2f __attribute__((ext_vector_type(2)));
typedef float v8f __attribute__((ext_vector_type(8)));

#define NROWS 131072   // B*T*NN = 128*128*8
#define EDGES 1048576

__device__ __forceinline__ v8f wmma_f32_k4(v2f a, v2f b, v8f c) {
  // D(16x16,f32) = A(16x4,f32) x B(4x16,f32) + C
  return __builtin_amdgcn_wmma_f32_16x16x4_f32(false, a, false, b, (short)0, c,
                                               false, false);
}

__device__ __forceinline__ void gatomic_add(float* p, float v) {
  (void)__hip_atomic_fetch_add(p, v, __ATOMIC_RELAXED, __HIP_MEMORY_SCOPE_AGENT);
}
__device__ __forceinline__ void satomic_add(float* p, float v) {
  (void)__hip_atomic_fetch_add(p, v, __ATOMIC_RELAXED, __HIP_MEMORY_SCOPE_WORKGROUP);
}

// ---------------- weight packing: B[kg][co] -> BP[(kg>>1)*128 + co*2 + (kg&1)]
// GCN weight W[K,64]: B[kg][co] = W[kg*64+co]
__global__ void pack_w_kernel(const float* __restrict__ W, float* __restrict__ BP) {
  int idx = blockIdx.x * blockDim.x + threadIdx.x;   // K*64
  int kg = idx >> 6, co = idx & 63;
  BP[((kg >> 1) << 7) + (co << 1) + (kg & 1)] = W[(kg << 6) + co];
}
// temporal conv weight tw[co,ci,1,3]: B[kg][co] = tw[co*192 + ci*3 + dt], kg = dt*64+ci
__global__ void pack_tw_kernel(const float* __restrict__ tw, float* __restrict__ BP) {
  int idx = blockIdx.x * blockDim.x + threadIdx.x;   // 192*64
  int kg = idx >> 6, co = idx & 63;
  int dt = kg >> 6, ci = kg & 63;
  BP[((kg >> 1) << 7) + (co << 1) + (kg & 1)] = tw[co * 192 + ci * 3 + dt];
}
// 1x1 conv weight ow[co,ci]: B[kg][co] = ow[co*64 + kg]
__global__ void pack_ow_kernel(const float* __restrict__ ow, float* __restrict__ BP) {
  int idx = blockIdx.x * blockDim.x + threadIdx.x;   // 64*64
  int kg = idx >> 6, co = idx & 63;
  BP[((kg >> 1) << 7) + (co << 1) + (kg & 1)] = ow[(co << 6) + kg];
}

// ---------------- degree / symmetric norm ----------------
__global__ void deg_init_kernel(float* __restrict__ deg) {
  int i = blockIdx.x * blockDim.x + threadIdx.x;
  deg[i] = 1.0f;                       // self loop
}
__global__ void deg_edge_kernel(const int* __restrict__ dst, float* __restrict__ deg) {
  int e = blockIdx.x * blockDim.x + threadIdx.x;
  gatomic_add(&deg[dst[e]], 1.0f);
}
__global__ void deg_fin_kernel(float* __restrict__ deg) {
  int i = blockIdx.x * blockDim.x + threadIdx.x;
  deg[i] = rsqrtf(deg[i]);             // in place: deg -> d^{-1/2}
}

// ---------------- generic GEMM: out[N,64] = A[N,K] @ B[K,64] (B packed) -----
// one wave computes a 16x64 tile; 4 accumulators share the A fragment.
__global__ __launch_bounds__(256) void gemm_wmma_kernel(
    const float* __restrict__ A, const float* __restrict__ BP,
    float* __restrict__ out, int K) {
  const int lane = threadIdx.x & 31;
  const int wave = (blockIdx.x * blockDim.x + threadIdx.x) >> 5;
  const int row0 = wave << 4;
  const int m  = lane & 15;            // A row / B col within tile
  const int kb = (lane >> 4) << 1;     // lanes 16-31 hold K+2,K+3
  const float* arow = A + (size_t)(row0 + m) * K;
  v8f acc0 = {}, acc1 = {}, acc2 = {}, acc3 = {};
  for (int k = 0; k < K; k += 4) {
    v2f a = *(const v2f*)(arow + k + kb);
    const float* bp = BP + (size_t)(((k + kb) >> 1) << 7) + (m << 1);
    v2f b0 = *(const v2f*)(bp);
    v2f b1 = *(const v2f*)(bp + 32);
    v2f b2 = *(const v2f*)(bp + 64);
    v2f b3 = *(const v2f*)(bp + 96);
    acc0 = wmma_f32_k4(a, b0, acc0);
    acc1 = wmma_f32_k4(a, b1, acc1);
    acc2 = wmma_f32_k4(a, b2, acc2);
    acc3 = wmma_f32_k4(a, b3, acc3);
  }
  const int mb = (lane >> 4) << 3;     // lanes 16-31 hold rows M+8
  v8f accs[4] = {acc0, acc1, acc2, acc3};
#pragma unroll
  for (int g = 0; g < 4; ++g) {
    const int co = (g << 4) + m;
#pragma unroll
    for (int j = 0; j < 8; ++j)
      out[(size_t)(row0 + mb + j) * 64 + co] = accs[g][j];
  }
}

// ---------------- graph aggregation ----------------
// Agg[i,c] = bias[c] + G[i,c]*dinv[i]^2   (self loop + bias)
__global__ void agg_init_kernel(const float* __restrict__ G,
                                const float* __restrict__ dinv,
                                const float* __restrict__ bias,
                                float* __restrict__ Agg) {
  size_t idx = (size_t)blockIdx.x * blockDim.x + threadIdx.x;  // N*64
  int row = (int)(idx >> 6), c = (int)(idx & 63);
  float di = dinv[row];
  Agg[idx] = bias[c] + G[idx] * di * di;
}
// Agg[d,c] += G[s,c] * dinv[s]*dinv[d]   (32 lanes per edge, float2/lane)
__global__ void agg_edge_kernel(const float* __restrict__ G,
                                const float* __restrict__ dinv,
                                const int* __restrict__ src,
                                const int* __restrict__ dst,
                                float* __restrict__ Agg) {
  size_t idx = (size_t)blockIdx.x * blockDim.x + threadIdx.x;  // E*32
  int e = (int)(idx >> 5);
  int c = (int)(idx & 31) << 1;
  int s = src[e], d = dst[e];
  float w = dinv[s] * dinv[d];
  v2f g = *(const v2f*)(G + ((size_t)s << 6) + c);
  float* ap = Agg + ((size_t)d << 6) + c;
  gatomic_add(ap, g.x * w);
  gatomic_add(ap + 1, g.y * w);
}

// ---------------- temporal (1,3) conv as implicit GEMM, K = 3*64 ------------
// rows of X are (b,t,n); accumulates per-channel sum/sumsq for BN.
__global__ __launch_bounds__(256) void tconv_wmma_kernel(
    const float* __restrict__ X, const float* __restrict__ BP,
    const float* __restrict__ tb, float* __restrict__ Y,
    float* __restrict__ bn_sum, float* __restrict__ bn_sq) {
  __shared__ float s_sum[64];
  __shared__ float s_sq[64];
  const int tid = threadIdx.x;
  if (tid < 64) { s_sum[tid] = 0.f; s_sq[tid] = 0.f; }
  __syncthreads();

  const int lane = tid & 31;
  const int wave = (blockIdx.x * blockDim.x + tid) >> 5;
  const int row0 = wave << 4;
  const int m  = lane & 15;
  const int kb = (lane >> 4) << 1;
  const int i  = row0 + m;
  const int b  = i >> 10;
  const int t  = (i >> 3) & 127;
  const int nn = i & 7;
  v8f acc0 = {}, acc1 = {}, acc2 = {}, acc3 = {};
  for (int k = 0; k < 192; k += 4) {
    const int kg = k + kb;             // kg,kg+1 stay inside one dt segment
    const int dt = kg >> 6;
    const int ci = kg & 63;
    const int ts = t + dt - 1;
    v2f a;
    if ((unsigned)ts < 128u) {
      a = *(const v2f*)(X + (size_t)((((b << 7) + ts) << 3) + nn) * 64 + ci);
    } else {
      a.x = 0.f; a.y = 0.f;            // zero padding in T
    }
    const float* bp = BP + (size_t)((kg >> 1) << 7) + (m << 1);
    v2f b0 = *(const v2f*)(bp);
    v2f b1 = *(const v2f*)(bp + 32);
    v2f b2 = *(const v2f*)(bp + 64);
    v2f b3 = *(const v2f*)(bp + 96);
    acc0 = wmma_f32_k4(a, b0, acc0);
    acc1 = wmma_f32_k4(a, b1, acc1);
    acc2 = wmma_f32_k4(a, b2, acc2);
    acc3 = wmma_f32_k4(a, b3, acc3);
  }
  const int mb = (lane >> 4) << 3;
  v8f accs[4] = {acc0, acc1, acc2, acc3};
#pragma unroll
  for (int g = 0; g < 4; ++g) {
    const int co = (g << 4) + m;
    const float bias = tb[co];
    float ps = 0.f, pq = 0.f;
#pragma unroll
    for (int j = 0; j < 8; ++j) {
      float v = accs[g][j] + bias;
      Y[(size_t)(row0 + mb + j) * 64 + co] = v;
      ps += v; pq += v * v;
    }
    satomic_add(&s_sum[co], ps);
    satomic_add(&s_sq[co], pq);
  }
  __syncthreads();
  if (tid < 64) {
    gatomic_add(&bn_sum[tid], s_sum[tid]);
    gatomic_add(&bn_sq[tid],  s_sq[tid]);
  }
}

// ---------------- BN stats -> affine (scale/shift) ----------------
__global__ void bn_fin_kernel(const float* __restrict__ bn_sum,
                              const float* __restrict__ bn_sq,
                              const float* __restrict__ g,
                              const float* __restrict__ beta,
                              float* __restrict__ scale,
                              float* __restrict__ shift) {
  int c = threadIdx.x;
  const float inv_n = 1.0f / (float)NROWS;
  float mu  = bn_sum[c] * inv_n;
  float var = bn_sq[c] * inv_n - mu * mu;   // biased variance (training mode)
  float rs  = rsqrtf(var + 1e-5f);
  float sc  = rs * g[c];
  scale[c] = sc;
  shift[c] = beta[c] - mu * sc;
}

__global__ void zero_kernel(float* __restrict__ p) { p[threadIdx.x] = 0.f; }

// ---------------- fused BN+ReLU -> 1x1 conv -> tanh, transposed store -------
// input rows (b,t,n); output written flat in (b,co,n,t) order = faithful
// view(-1, C) layout the next layer consumes directly.
__global__ __launch_bounds__(256) void onexone_wmma_kernel(
    const float* __restrict__ Y, const float* __restrict__ BP,
    const float* __restrict__ ob, const float* __restrict__ scale,
    const float* __restrict__ shift, float* __restrict__ out) {
  const int lane = threadIdx.x & 31;
  const int wave = (blockIdx.x * blockDim.x + threadIdx.x) >> 5;
  const int row0 = wave << 4;
  const int m  = lane & 15;
  const int kb = (lane >> 4) << 1;
  const float* yrow = Y + (size_t)(row0 + m) * 64;
  v8f acc0 = {}, acc1 = {}, acc2 = {}, acc3 = {};
  for (int k = 0; k < 64; k += 4) {
    const int kg = k + kb;
    v2f y = *(const v2f*)(yrow + kg);
    v2f a;
    a.x = fmaxf(y.x * scale[kg]     + shift[kg],     0.f);
    a.y = fmaxf(y.y * scale[kg + 1] + shift[kg + 1], 0.f);
    const float* bp = BP + (size_t)((kg >> 1) << 7) + (m << 1);
    v2f b0 = *(const v2f*)(bp);
    v2f b1 = *(const v2f*)(bp + 32);
    v2f b2 = *(const v2f*)(bp + 64);
    v2f b3 = *(const v2f*)(bp + 96);
    acc0 = wmma_f32_k4(a, b0, acc0);
    acc1 = wmma_f32_k4(a, b1, acc1);
    acc2 = wmma_f32_k4(a, b2, acc2);
    acc3 = wmma_f32_k4(a, b3, acc3);
  }
  const int mb = (lane >> 4) << 3;
  v8f accs[4] = {acc0, acc1, acc2, acc3};
#pragma unroll
  for (int g = 0; g < 4; ++g) {
    const int co = (g << 4) + m;
    const float bias = ob[co];
#pragma unroll
    for (int j = 0; j < 8; ++j) {
      const int r  = row0 + mb + j;
      const int bb = r >> 10;
      const int t  = (r >> 3) & 127;
      const int nn = r & 7;
      out[(size_t)((((bb << 6) + co) << 3) + nn) * 128 + t] =
          tanhf(accs[g][j] + bias);
    }
  }
}

// ---------------- head: (H @ reg_w + reg_b) then mean over (T, feat) --------
// mean o matmul == dot with column-summed weights
__global__ void head_kernel(const float* __restrict__ H,
                            const float* __restrict__ rw,
                            const float* __restrict__ rb,
                            float* __restrict__ out) {
  __shared__ float wsum[64];
  __shared__ float red[128];
  const int tid = threadIdx.x;   // = t
  if (tid < 64) {
    float s = 0.f;
#pragma unroll
    for (int f = 0; f < 8; ++f) s += rw[tid * 8 + f];
    wsum[tid] = s;
  }
  __syncthreads();
  const int b = blockIdx.x >> 3, n = blockIdx.x & 7;
  const int i = ((b << 7) + tid) * 8 + n;      // row (b,t,n)
  const float* hp = H + (size_t)i * 64;
  float dot = 0.f;
#pragma unroll 8
  for (int c = 0; c < 64; ++c) dot += hp[c] * wsum[c];
  red[tid] = dot;
  __syncthreads();
  for (int s = 64; s > 0; s >>= 1) {
    if (tid < s) red[tid] += red[tid + s];
    __syncthreads();
  }
  if (tid == 0) {
    float rbs = 0.f;
#pragma unroll
    for (int f = 0; f < 8; ++f) rbs += rb[f];
    out[blockIdx.x] = red[0] * (1.0f / 1024.0f) + rbs * (1.0f / 8.0f);
  }
}

extern "C" void kernel_launch(void* const* d_in, const int* in_sizes, int n_in,
                              void* d_out, int out_size, void* d_ws, size_t ws_size,
                              hipStream_t stream) {
  const float* x      = (const float*)d_in[0];
  const int*   ei     = (const int*)d_in[1];
  const float* gcn1_w = (const float*)d_in[2];
  const float* gcn1_b = (const float*)d_in[3];
  const float* t1_w   = (const float*)d_in[4];
  const float* t1_b   = (const float*)d_in[5];
  const float* bn1_g  = (const float*)d_in[6];
  const float* bn1_b  = (const float*)d_in[7];
  const float* o1_w   = (const float*)d_in[8];
  const float* o1_b   = (const float*)d_in[9];
  const float* gcn2_w = (const float*)d_in[10];
  const float* gcn2_b = (const float*)d_in[11];
  const float* t2_w   = (const float*)d_in[12];
  const float* t2_b   = (const float*)d_in[13];
  const float* bn2_g  = (const float*)d_in[14];
  const float* bn2_b  = (const float*)d_in[15];
  const float* o2_w   = (const float*)d_in[16];
  const float* o2_b   = (const float*)d_in[17];
  const float* reg_w  = (const float*)d_in[18];
  const float* reg_b  = (const float*)d_in[19];
  float* out = (float*)d_out;

  const int* srcp = ei;
  const int* dstp = ei + EDGES;

  // workspace layout
  float* P0    = (float*)d_ws;                      // N*64
  float* P1    = P0 + (size_t)NROWS * 64;           // N*64
  float* P2    = P1 + (size_t)NROWS * 64;           // N*64
  float* dinv  = P2 + (size_t)NROWS * 64;           // N
  float* bnbuf = dinv + NROWS;                      // 128 (sum + sumsq)
  float* scl   = bnbuf + 128;                       // 64
  float* shf   = scl + 64;                          // 64
  float* WPg   = shf + 64;                          // 64*64  packed GCN W
  float* TWP   = WPg + 64 * 64;                     // 192*64 packed temporal W
  float* OWP   = TWP + 192 * 64;                    // 64*64  packed 1x1 W

  const int gemm_blocks = NROWS / 128;              // 8 waves * 16 rows / block

  // degrees (shared by both layers): dinv = (deg+1)^{-1/2}
  deg_init_kernel<<<NROWS / 256, 256, 0, stream>>>(dinv);
  deg_edge_kernel<<<EDGES / 256, 256, 0, stream>>>(dstp, dinv);
  deg_fin_kernel<<<NROWS / 256, 256, 0, stream>>>(dinv);

  // ---- layer 1 ----
  pack_w_kernel <<<(32 * 64) / 256, 256, 0, stream>>>(gcn1_w, WPg);
  pack_tw_kernel<<<(192 * 64) / 256, 256, 0, stream>>>(t1_w, TWP);
  pack_ow_kernel<<<(64 * 64) / 256, 256, 0, stream>>>(o1_w, OWP);
  gemm_wmma_kernel<<<gemm_blocks, 256, 0, stream>>>(x, WPg, P0, 32);
  agg_init_kernel<<<(NROWS * 64) / 256, 256, 0, stream>>>(P0, dinv, gcn1_b, P1);
  agg_edge_kernel<<<(EDGES * 32) / 256, 256, 0, stream>>>(P0, dinv, srcp, dstp, P1);
  zero_kernel<<<1, 128, 0, stream>>>(bnbuf);
  tconv_wmma_kernel<<<gemm_blocks, 256, 0, stream>>>(P1, TWP, t1_b, P0, bnbuf, bnbuf + 64);
  bn_fin_kernel<<<1, 64, 0, stream>>>(bnbuf, bnbuf + 64, bn1_g, bn1_b, scl, shf);
  onexone_wmma_kernel<<<gemm_blocks, 256, 0, stream>>>(P0, OWP, o1_b, scl, shf, P2);

  // ---- layer 2 (P2 reinterpreted as [N,64], faithful to view(-1, C)) ----
  pack_w_kernel <<<(64 * 64) / 256, 256, 0, stream>>>(gcn2_w, WPg);
  pack_tw_kernel<<<(192 * 64) / 256, 256, 0, stream>>>(t2_w, TWP);
  pack_ow_kernel<<<(64 * 64) / 256, 256, 0, stream>>>(o2_w, OWP);
  gemm_wmma_kernel<<<gemm_blocks, 256, 0, stream>>>(P2, WPg, P0, 64);
  agg_init_kernel<<<(NROWS * 64) / 256, 256, 0, stream>>>(P0, dinv, gcn2_b, P1);
  agg_edge_kernel<<<(EDGES * 32) / 256, 256, 0, stream>>>(P0, dinv, srcp, dstp, P1);
  zero_kernel<<<1, 128, 0, stream>>>(bnbuf);
  tconv_wmma_kernel<<<gemm_blocks, 256, 0, stream>>>(P1, TWP, t2_b, P0, bnbuf, bnbuf + 64);
  bn_fin_kernel<<<1, 64, 0, stream>>>(bnbuf, bnbuf + 64, bn2_g, bn2_b, scl, shf);
  onexone_wmma_kernel<<<gemm_blocks, 256, 0, stream>>>(P0, OWP, o2_b, scl, shf, P2);

  // ---- head ----
  head_kernel<<<128 * 8, 128, 0, stream>>>(P2, reg_w, reg_b, out);
}